// HopfieldMemory_26749056319692
// MI455X (gfx1250) — compile-verified
//
#include <hip/hip_runtime.h>
#include <hip/hip_bf16.h>

typedef float v2f __attribute__((ext_vector_type(2)));
typedef float v8f __attribute__((ext_vector_type(8)));

#define B_  8
#define S_  2048
#define D_  1024
#define M_  4096
#define d_  256
#define DECAY 0.95f

// ---------------------------------------------------------------------------
// 1) Partial sums over S for x_summary.  x:[B,S,D].  Each block sums a chunk
//    of 64 rows of S for 256 consecutive d-columns.  Fully coalesced.
//    part layout: [sc=32][b=8][D]
// ---------------------------------------------------------------------------
__global__ void k_xsum_partial(const float* __restrict__ x, float* __restrict__ part) {
    int dcol = blockIdx.x * 256 + threadIdx.x;
    int b    = blockIdx.y;
    int sc   = blockIdx.z;
    const float* p = x + ((size_t)b * S_ + (size_t)sc * 64) * D_ + dcol;
    float s = 0.f;
#pragma unroll 4
    for (int i = 0; i < 64; ++i) s += p[(size_t)i * D_];
    part[((size_t)sc * B_ + b) * D_ + dcol] = s;
}

// Reduce the 32 partials -> mean.  xs:[16,D] (rows 8..15 are pad, never read
// as meaningful data -- only to keep WMMA A-loads in-bounds).
__global__ void k_xsum_reduce(const float* __restrict__ part, float* __restrict__ xs) {
    int i = blockIdx.x * 256 + threadIdx.x;   // 0 .. B*D-1
    float s = 0.f;
#pragma unroll 8
    for (int sc = 0; sc < 32; ++sc) s += part[(size_t)sc * (B_ * D_) + i];
    xs[i] = s * (1.0f / (float)S_);
}

// ---------------------------------------------------------------------------
// 2) Padded-M=16 WMMA GEMM:  C[16 x N] = A[16 x K] * B[K x N] (+bias)
//    All buffers are 16 rows; rows 8..15 carry don't-care data (WMMA rows are
//    independent, so pad rows never pollute rows 0..7).  No predication.
//    One wave per 16-wide N tile, 4 waves per block.
//    V_WMMA_F32_16X16X4_F32 layouts (ISA 7.12.2):
//      A 16x4:  lane<16 -> A[r][k0..k0+1]; lane>=16 -> A[r][k0+2..k0+3]
//      B 4x16:  rows striped like C/D: lanes 0-15 rows k0,k0+1; 16-31 k0+2,k0+3
//      D:       VGPR j -> row j (lanes 0-15) / j+8 (lanes 16-31), col lane&15
//    BTRANS=1 reads B[k][n] = Bsrc[n*ldb + k].
// ---------------------------------------------------------------------------
template <int BTRANS>
__global__ void k_wmma_gemm(const float* __restrict__ A, int lda,
                            const float* __restrict__ B, int ldb,
                            const float* __restrict__ bias,
                            float* __restrict__ C, int ldc, int K) {
    const int wave = threadIdx.x >> 5;                 // 4 waves / block
    const int n0   = (blockIdx.x * 4 + wave) * 16;
    const int lane = threadIdx.x & 31;
    const int half = lane >> 4;
    const int r    = lane & 15;
    const int n    = n0 + r;

    const float* pA = A + (size_t)r * lda + half * 2;
    const float* pB = BTRANS ? (B + (size_t)n * ldb + half * 2)
                             : (B + (size_t)(half * 2) * ldb + n);

    v8f acc = {0.f, 0.f, 0.f, 0.f, 0.f, 0.f, 0.f, 0.f};

    for (int k0 = 0; k0 < K; k0 += 8) {                // 2 WMMAs per iter
        v2f a0, a1, b0, b1;
        a0.x = pA[0]; a0.y = pA[1];                    // b64 load
        a1.x = pA[4]; a1.y = pA[5];                    // b64 load
        if (BTRANS) {
            b0.x = pB[0]; b0.y = pB[1];
            b1.x = pB[4]; b1.y = pB[5];
        } else {
            b0.x = pB[0];                 b0.y = pB[(size_t)ldb];
            b1.x = pB[(size_t)4 * ldb];   b1.y = pB[(size_t)5 * ldb];
        }
        acc = __builtin_amdgcn_wmma_f32_16x16x4_f32(
            false, a0, false, b0, (short)0, acc, false, false);
        acc = __builtin_amdgcn_wmma_f32_16x16x4_f32(
            false, a1, false, b1, (short)0, acc, false, false);
        pA += 8;
        pB += BTRANS ? 8 : (size_t)8 * ldb;
    }

    const float bv = bias ? bias[n] : 0.f;
#pragma unroll
    for (int j = 0; j < 8; ++j)
        C[(size_t)(j + half * 8) * ldc + n] = acc[j] + bv;
}

// ---------------------------------------------------------------------------
// 3) write gate: one wave per batch. gate[b] = sigmoid(xs[b]·Wg + bg)
// ---------------------------------------------------------------------------
__global__ void k_gate(const float* __restrict__ xs, const float* __restrict__ Wg,
                       const float* __restrict__ bg, float* __restrict__ gate) {
    int b = threadIdx.y;       // block (32, 8)
    int lane = threadIdx.x;
    float s = 0.f;
    for (int k = lane; k < D_; k += 32) s += xs[(size_t)b * D_ + k] * Wg[k];
    for (int off = 16; off; off >>= 1) s += __shfl_down(s, off, 32);
    if (lane == 0) gate[b] = 1.0f / (1.0f + __expf(-(s + bg[0])));
}

// ---------------------------------------------------------------------------
// 4) In-place softmax over a row of 4096 (one block per batch row 0..7)
// ---------------------------------------------------------------------------
__global__ void k_softmax4096(float* __restrict__ s) {
    float* row = s + (size_t)blockIdx.x * M_;
    __shared__ float red[256];
    const int t = threadIdx.x;

    float m = -3.4e38f;
    for (int i = t; i < M_; i += 256) m = fmaxf(m, row[i]);
    red[t] = m; __syncthreads();
    for (int o = 128; o; o >>= 1) { if (t < o) red[t] = fmaxf(red[t], red[t + o]); __syncthreads(); }
    m = red[0]; __syncthreads();

    float sum = 0.f;
    for (int i = t; i < M_; i += 256) { float e = __expf(row[i] - m); row[i] = e; sum += e; }
    red[t] = sum; __syncthreads();
    for (int o = 128; o; o >>= 1) { if (t < o) red[t] += red[t + o]; __syncthreads(); }
    const float inv = 1.0f / red[0];
    __syncthreads();
    for (int i = t; i < M_; i += 256) row[i] *= inv;
}

// ---------------------------------------------------------------------------
// 5) w_scores[b][m] = query[b] · memory[b][m]   (per-batch memory, GEMV)
// ---------------------------------------------------------------------------
__global__ void k_wscores(const float* __restrict__ query, const float* __restrict__ mem,
                          float* __restrict__ ws) {
    const int idx = blockIdx.x * 256 + threadIdx.x;   // b*M + m
    const int b = idx >> 12;
    const float4* qp = (const float4*)(query + (size_t)b * d_);
    const float4* mp = (const float4*)(mem + (size_t)idx * d_);
    __builtin_prefetch(mp + 64, 0, 1);   // next row -> global_prefetch_b8
    float s = 0.f;
#pragma unroll 8
    for (int i = 0; i < d_ / 4; ++i) {
        float4 q = qp[i], v = mp[i];
        s += q.x * v.x + q.y * v.y + q.z * v.z + q.w * v.w;
    }
    ws[idx] = s;
}

// ---------------------------------------------------------------------------
// 6) x_augmented = x + read_out[b] broadcast over seq.  float4 streaming.
// ---------------------------------------------------------------------------
__global__ void k_xaug(const float* __restrict__ x, const float* __restrict__ ro,
                       float* __restrict__ out) {
    const size_t i = (size_t)blockIdx.x * 256 + threadIdx.x;  // float4 index
    float4 v = ((const float4*)x)[i];
    const size_t f = i * 4;
    const int dcol = (int)(f & (D_ - 1));
    const int b    = (int)(f >> 21);                   // S_*D_ = 2^21 floats/batch
    const float* r = ro + (size_t)b * D_ + dcol;
    v.x += r[0]; v.y += r[1]; v.z += r[2]; v.w += r[3];
    ((float4*)out)[i] = v;
}

// ---------------------------------------------------------------------------
// 7) memory_new = DECAY*memory + (1-DECAY) * w_attn[b,m]*gate[b]*write_value[b,:]
// ---------------------------------------------------------------------------
__global__ void k_memupd(const float* __restrict__ mem, const float* __restrict__ wattn,
                         const float* __restrict__ gate, const float* __restrict__ wv,
                         float* __restrict__ out) {
    const size_t i = (size_t)blockIdx.x * 256 + threadIdx.x;  // float4 index
    const size_t f = i * 4;
    const int    k  = (int)(f & (d_ - 1));
    const size_t bm = f >> 8;                                 // b*M + m
    const int    b  = (int)(bm >> 12);
    const float  w  = wattn[bm] * gate[b] * (1.0f - DECAY);
    const float4 v   = ((const float4*)mem)[i];
    const float4 wvv = *(const float4*)(wv + (size_t)b * d_ + k);
    float4 o;
    o.x = DECAY * v.x + w * wvv.x;
    o.y = DECAY * v.y + w * wvv.y;
    o.z = DECAY * v.z + w * wvv.z;
    o.w = DECAY * v.w + w * wvv.w;
    ((float4*)out)[i] = o;
}

// ---------------------------------------------------------------------------
extern "C" void kernel_launch(void* const* d_in, const int* in_sizes, int n_in,
                              void* d_out, int out_size, void* d_ws, size_t ws_size,
                              hipStream_t stream) {
    const float* x   = (const float*)d_in[0];
    const float* mem = (const float*)d_in[1];
    const float* Wq  = (const float*)d_in[2];
    const float* bq  = (const float*)d_in[3];
    const float* Wv  = (const float*)d_in[4];
    const float* bv  = (const float*)d_in[5];
    const float* Wo  = (const float*)d_in[6];
    const float* bo  = (const float*)d_in[7];
    const float* Wg  = (const float*)d_in[8];
    const float* bg  = (const float*)d_in[9];

    float* out_xaug = (float*)d_out;                                   // [B,S,D]
    float* out_mem  = out_xaug + (size_t)B_ * S_ * D_;                 // [B,M,d]

    // workspace carve-up (floats).  All WMMA A/C-side buffers are 16 rows;
    // rows 8..15 hold don't-care data that is computed but never consumed.
    float* w = (float*)d_ws;
    float* part  = w;               // 32*8*1024 = 262144
    float* xs    = part + 262144;   // 16*1024  = 16384
    float* query = xs + 16384;      // 16*256   = 4096
    float* wval  = query + 4096;    // 16*256   = 4096
    float* gate  = wval + 4096;     // 16
    float* rs    = gate + 16;       // 16*4096  = 65536 (read scores -> attn)
    float* wsc   = rs + 65536;      // 8*4096   = 32768 (write scores -> attn)
    float* retr  = wsc + 32768;     // 16*256   = 4096
    float* rout  = retr + 4096;     // 16*1024  = 16384
    (void)ws_size; (void)in_sizes; (void)n_in; (void)out_size;

    // --- x_summary (deterministic two-stage mean over S) ---
    k_xsum_partial<<<dim3(D_ / 256, B_, 32), 256, 0, stream>>>(x, part);
    k_xsum_reduce<<<dim3((B_ * D_) / 256), 256, 0, stream>>>(part, xs);

    // --- query = xs @ Wq + bq ; write_value = xs @ Wv + bv  (WMMA) ---
    k_wmma_gemm<0><<<dim3(d_ / 16 / 4), 128, 0, stream>>>(xs, D_, Wq, d_, bq, query, d_, D_);
    k_wmma_gemm<0><<<dim3(d_ / 16 / 4), 128, 0, stream>>>(xs, D_, Wv, d_, bv, wval,  d_, D_);

    // --- write gate ---
    k_gate<<<1, dim3(32, B_), 0, stream>>>(xs, Wg, bg, gate);

    // --- read path: scores = query @ patterns^T -> softmax -> retrieved -> read_out ---
    const float* patterns = mem;  // memory[0] : [M, d]
    k_wmma_gemm<1><<<dim3(M_ / 16 / 4), 128, 0, stream>>>(query, d_, patterns, d_, nullptr, rs, M_, d_);
    k_softmax4096<<<dim3(B_), 256, 0, stream>>>(rs);
    k_wmma_gemm<0><<<dim3(d_ / 16 / 4), 128, 0, stream>>>(rs, M_, patterns, d_, nullptr, retr, d_, M_);
    k_wmma_gemm<0><<<dim3(D_ / 16 / 4), 128, 0, stream>>>(retr, d_, Wo, D_, bo, rout, D_, d_);

    // --- write path: per-batch scores -> softmax ---
    k_wscores<<<dim3((B_ * M_) / 256), 256, 0, stream>>>(query, mem, wsc);
    k_softmax4096<<<dim3(B_), 256, 0, stream>>>(wsc);

    // --- big streaming outputs ---
    k_xaug<<<dim3((B_ * S_ * D_) / 4 / 256), 256, 0, stream>>>(x, rout, out_xaug);
    k_memupd<<<dim3((B_ * M_ * d_) / 4 / 256), 256, 0, stream>>>(mem, wsc, gate, wval, out_mem);
}